// And_56281251447399
// MI455X (gfx1250) — compile-verified
//
#include <hip/hip_runtime.h>
#include <hip/hip_bf16.h>
#include <stdint.h>

// Tropical (max-min) matmul: out[b,o] = max_i min(m[b,i], clamp(w[i,o],0,1))
// B=1024, IN=512, OUT=256, all f32. VALU-bound (min/max semiring -> no WMMA
// instruction applies); CDNA5 features: global_load_async_to_lds_b32 with
// per-lane LDS scatter (transposing copy, ASYNCcnt) + s_wait_asynccnt,
// double-buffered LDS with compile-time buffer selection, ds_load_b128
// feeding 4x4 register micro-tiles (32 VALU ops per 2 DS loads per thread).

typedef __attribute__((ext_vector_type(4))) float v4f;

#define B_DIM   1024
#define IN_DIM  512
#define OUT_DIM 256

#define BM 32           // b-tile per block
#define BN 32           // o-tile per block
#define BK 32           // reduction chunk
#define NCHUNK (IN_DIM / BK)   // 16 (even -> clean unroll-by-2)
#define SMS 36          // padded LDS row stride for m tile (rows 16B-aligned)
#define NTHR 64         // 2 waves; 8x8 thread grid, 4x4 micro-tile each

// Low 32 bits of a generic pointer into __shared__ == LDS byte offset
// (flat-LDS aperture keeps the workgroup-relative offset in addr[31:0]).
__device__ __forceinline__ uint32_t lds_byte_off(const void* p) {
  return (uint32_t)(uintptr_t)p;
}

// Async global->LDS copy, 4 bytes per lane, per-lane LDS dest address.
// Tracked by ASYNCcnt; EXEC is all-ones here (no divergence in this kernel).
__device__ __forceinline__ void async_g2l_b32(uint32_t lds_addr, const float* gaddr) {
  asm volatile("global_load_async_to_lds_b32 %0, %1, off"
               :: "v"(lds_addr), "v"(gaddr)
               : "memory");
}

__device__ __forceinline__ void wait_asynccnt0() {
  asm volatile("s_wait_asynccnt 0" ::: "memory");
}

__global__ __launch_bounds__(NTHR)
void minmax_mm_kernel(const float* __restrict__ m,
                      const float* __restrict__ w,
                      float* __restrict__ out) {
  __shared__ __align__(16) float sm[2][BK][SMS];  // m chunk, [i][b] (transposed)
  __shared__ __align__(16) float sw[2][BK][BN];   // clamped w chunk, [i][o]

  const int tid = threadIdx.x;
  const int b0  = blockIdx.x * BM;
  const int o0  = blockIdx.y * BN;

  // compute-phase coordinates: 8 threads along b, 8 along o, 4x4 micro-tile
  const int tb = (tid & 7) * 4;
  const int to = (tid >> 3) * 4;

  // ---- m staging decomposition (transposing async copy) ----
  // element e = tid + 64*r  ->  i = tid&31 (lane-contiguous global reads),
  //                             b = tid>>5 + 2*r  (LDS scatter along rows)
  const int mi  = tid & 31;
  const int mb0 = tid >> 5;
  const float* m_gbase = m + (size_t)(b0 + mb0) * IN_DIM + mi;
  const uint32_t sm_base[2] = { lds_byte_off(&sm[0][mi][mb0]),
                                lds_byte_off(&sm[1][mi][mb0]) };

  // ---- w staging decomposition (global->regs->clamp->LDS) ----
  const int wc4 = (tid & 7) * 4;   // column (float4) within row
  const int wr0 = tid >> 3;        // row base; rows wr0 + 8*r
  v4f wreg[4];

  // issue the transposing async copy of one m chunk into buffer `sbuf`
  auto issue_m = [&](int kk, uint32_t lbase) {
#pragma unroll
    for (int r = 0; r < 16; ++r)
      async_g2l_b32(lbase + r * 8u, m_gbase + kk + (size_t)(2 * r) * IN_DIM);
  };
  auto load_w = [&](int kk) {
#pragma unroll
    for (int r = 0; r < 4; ++r)
      wreg[r] = *(const v4f*)(w + (size_t)(kk + wr0 + 8 * r) * OUT_DIM + o0 + wc4);
  };

  float acc[4][4];
#pragma unroll
  for (int x = 0; x < 4; ++x)
#pragma unroll
    for (int y = 0; y < 4; ++y)
      acc[x][y] = 0.0f;   // min(m,w) >= 0 always (m in [0,1], w clamped)

  // clamp + commit prefetched w tile into buffer `sbuf`
  auto store_w = [&](int sbuf) {
#pragma unroll
    for (int r = 0; r < 4; ++r) {
      v4f t = wreg[r];
#pragma unroll
      for (int c = 0; c < 4; ++c) t[c] = fminf(fmaxf(t[c], 0.0f), 1.0f);
      *(v4f*)&sw[sbuf][wr0 + 8 * r][wc4] = t;
    }
  };

  // compute one chunk from buffer `sbuf`: 2x ds_load_b128 -> 16 min + 16 max
  auto compute = [&](int sbuf) {
#pragma unroll 8
    for (int i = 0; i < BK; ++i) {
      v4f mv = *(const v4f*)&sm[sbuf][i][tb];
      v4f wv = *(const v4f*)&sw[sbuf][i][to];
#pragma unroll
      for (int x = 0; x < 4; ++x)
#pragma unroll
        for (int y = 0; y < 4; ++y)
          acc[x][y] = fmaxf(acc[x][y], fminf(mv[x], wv[y]));
    }
  };

  // ---------------- prologue: stage chunk 0 into buffer 0 ----------------
  issue_m(0, sm_base[0]);
  load_w(0);
  store_w(0);
  wait_asynccnt0();
  __syncthreads();

  // ------------- main pipelined loop, unrolled x2: static buffers -------------
  for (int k = 0; k < NCHUNK; k += 2) {
    // --- even stage: compute buf0, prefetch k+1 into buf1 ---
    issue_m((k + 1) * BK, sm_base[1]);
    load_w((k + 1) * BK);
    compute(0);
    store_w(1);
    wait_asynccnt0();           // buf1 m-tile resident before barrier release
    __syncthreads();

    // --- odd stage: compute buf1, prefetch k+2 into buf0 ---
    if (k + 2 < NCHUNK) {
      issue_m((k + 2) * BK, sm_base[0]);
      load_w((k + 2) * BK);
    }
    compute(1);
    if (k + 2 < NCHUNK) {
      store_w(0);
      wait_asynccnt0();
      __syncthreads();
    }
  }

  // ---------------- epilogue: 4 x global_store_b128 ----------------
#pragma unroll
  for (int x = 0; x < 4; ++x) {
    v4f o;
#pragma unroll
    for (int y = 0; y < 4; ++y) o[y] = acc[x][y];
    *(v4f*)(out + (size_t)(b0 + tb + x) * OUT_DIM + o0 + to) = o;
  }
}

extern "C" void kernel_launch(void* const* d_in, const int* in_sizes, int n_in,
                              void* d_out, int out_size, void* d_ws, size_t ws_size,
                              hipStream_t stream) {
  (void)in_sizes; (void)n_in; (void)out_size; (void)d_ws; (void)ws_size;
  const float* m = (const float*)d_in[0];     // [1024, 512]
  const float* w = (const float*)d_in[1];     // [512, 256]
  float* out = (float*)d_out;                 // [1024, 256]

  dim3 grid(B_DIM / BM, OUT_DIM / BN);        // 32 x 8 = 256 blocks
  minmax_mm_kernel<<<grid, NTHR, 0, stream>>>(m, w, out);
}